// MovingAvg_76510547411146
// MI455X (gfx1250) — compile-verified
//
#include <hip/hip_runtime.h>
#include <hip/hip_bf16.h>

// ---------------------------------------------------------------------------
// out[b,k,0,m,n] = sum_t W[k,t] * x[b,t,0,m,n]
// W (12x24) folds the 12-step EMA scan into one matrix, computed at compile
// time. fp32 WMMA (V_WMMA_F32_16X16X4_F32) keeps exact fp32 precision while
// offloading the 288 MAC/element to the matrix core -> kernel stays HBM-bound
// (~38 MB total traffic, ~1.6 us at 23.3 TB/s).
// ---------------------------------------------------------------------------

#define T_IN   24
#define T_OUT  12
#define D_DIM  10
#define MN     8192          // 64*128
#define B_DIM  32
#define T_STRIDE 81920       // D_DIM * MN (t stride in floats)
#define B_STRIDE_IN  1966080 // T_IN * T_STRIDE
#define B_STRIDE_OUT 98304   // T_OUT * MN
#define GROUPS_PER_B 512     // MN / 16
#define N_GROUPS 16384       // B_DIM * GROUPS_PER_B

typedef __attribute__((ext_vector_type(2))) float v2f;
typedef __attribute__((ext_vector_type(8))) float v8f;

struct WMat { float w[16][T_IN]; };   // rows 12..15 stay zero (WMMA M padding)

static constexpr WMat make_w() {
    WMat R{};
    constexpr double MU = 2.0 / (T_IN + 1);
    double c[T_IN] = {};
    double p = 1.0;
    for (int i = 0; i < T_IN - 1; ++i) p *= (1.0 - MU);
    const double base = p / T_IN;
    for (int j = 0; j < T_IN; ++j) {
        double rec = 0.0;
        if (j < T_IN - 1) {
            double q = 1.0;
            for (int i = 0; i < T_IN - 2 - j; ++i) q *= (1.0 - MU);
            rec = MU * q;
        }
        c[j] = base + rec;
    }
    // Unroll the scan: p_k = sum_t c[t]*e[k+t], e = (s[0..23], p_0, p_1, ...)
    double Wd[T_OUT][T_IN] = {};
    for (int k = 0; k < T_OUT; ++k) {
        for (int i = 0; i < T_IN; ++i) {
            double v = (i >= k) ? c[i - k] : 0.0;
            for (int m = 0; m < k; ++m) v += c[T_IN - k + m] * Wd[m][i];
            Wd[k][i] = v;
        }
    }
    for (int k = 0; k < T_OUT; ++k)
        for (int i = 0; i < T_IN; ++i)
            R.w[k][i] = (float)Wd[k][i];
    return R;
}

__constant__ WMat W_const = make_w();

__global__ void __launch_bounds__(256)
MovingAvg_76510547411146_kernel(const float* __restrict__ x,
                                float* __restrict__ out) {
    const int lane = threadIdx.x & 31;
    const int wave = threadIdx.x >> 5;
    const int g    = blockIdx.x * 8 + wave;          // one 16-element group / wave

    const int b    = g / GROUPS_PER_B;
    const int mnb  = (g % GROUPS_PER_B) * 16;
    const int col  = lane & 15;                      // N (element) / M (W row)
    const int half = lane >> 4;

    // ---- A matrix: W K-chunks, resident for the whole wave --------------
    // A 16x4 f32 layout: lane L -> row M=L%16; VGPR v -> K = 2*half + v.
    v2f A[6];
#pragma unroll
    for (int kc = 0; kc < 6; ++kc) {
        const int t0 = kc * 4 + 2 * half;
        A[kc].x = W_const.w[col][t0];
        A[kc].y = W_const.w[col][t0 + 1];
    }

    const float* __restrict__ xb = x + (size_t)b * B_STRIDE_IN + mnb;

    // ---- 6 x WMMA accumulate over K = 24 --------------------------------
    // B 4x16 f32 layout: VGPR v, lane-half h -> K = v + 2*h, N = lane%16.
    v8f acc = {};
#pragma unroll
    for (int kc = 0; kc < 6; ++kc) {
        const int t0 = kc * 4 + 2 * half;
        v2f Bv;
        Bv.x = xb[(size_t)t0 * T_STRIDE + col];
        Bv.y = xb[(size_t)(t0 + 1) * T_STRIDE + col];
        acc = __builtin_amdgcn_wmma_f32_16x16x4_f32(
            /*neg_a=*/false, A[kc], /*neg_b=*/false, Bv,
            /*c_mod=*/(short)0, acc, /*reuse_a=*/false, /*reuse_b=*/false);
    }

    // ---- Store D rows 0..11 (rows 12..15 are the zero-padding) ----------
    // D layout: VGPR r, lane-half h -> row M = r + 8*h, col N = lane%16.
    // Lane-half 0 owns k = 0..7, lane-half 1 owns k = 8..11 (+ 4 pad rows).
    float* __restrict__ op =
        out + (size_t)b * B_STRIDE_OUT + (size_t)(8 * half) * MN + mnb + col;
#pragma unroll
    for (int r = 0; r < 4; ++r)            // valid for both halves
        op[(size_t)r * MN] = acc[r];
    if (half == 0) {                       // single branch for the 4 tail rows
#pragma unroll
        for (int r = 4; r < 8; ++r)
            op[(size_t)r * MN] = acc[r];
    }
}

extern "C" void kernel_launch(void* const* d_in, const int* in_sizes, int n_in,
                              void* d_out, int out_size, void* d_ws, size_t ws_size,
                              hipStream_t stream) {
    (void)in_sizes; (void)n_in; (void)out_size; (void)d_ws; (void)ws_size;
    const float* x = (const float*)d_in[0];
    float* out = (float*)d_out;
    // 16384 waves, 8 waves (256 threads) per block -> 2048 blocks.
    MovingAvg_76510547411146_kernel<<<dim3(N_GROUPS / 8), dim3(256), 0, stream>>>(x, out);
}